// LightGCN_12197707121044
// MI455X (gfx1250) — compile-verified
//
#include <hip/hip_runtime.h>
#include <stdint.h>

typedef unsigned int u32;
typedef u32 u32x4 __attribute__((ext_vector_type(4)));
typedef int  i32x8 __attribute__((ext_vector_type(8)));
typedef int  i32x4 __attribute__((ext_vector_type(4)));

#define LGC_CHUNK 1024
#define SCATTER_BLOCK 256

// ---------------------------------------------------------------------------
// TDM: DMA a contiguous run of 4-byte elements from global memory into LDS.
// D# layout per CDNA5 ISA:
//   group0: count=1, lds_addr, global_addr[56:0], type=2
//   group1: data_size=4B, tensor_dim0=remaining (OOB reads -> 0 for the tail),
//           tensor_dim1=1, tile_dim0=tile_elems, tile_dim1=1
// Tracked with TENSORcnt; caller waits with s_wait_tensorcnt.
// ---------------------------------------------------------------------------
__device__ __forceinline__ void tdm_load_1d(u32 lds_byte_addr, const void* gsrc,
                                            u32 remaining_elems, u32 tile_elems) {
  const unsigned long long ga = (unsigned long long)gsrc;
  u32x4 g0;
  g0[0] = 1u;                                            // count=1 (valid user D#)
  g0[1] = lds_byte_addr;                                 // lds_addr (bytes)
  g0[2] = (u32)(ga & 0xFFFFFFFFull);                     // global_addr[31:0]
  g0[3] = (u32)((ga >> 32) & 0x1FFFFFFull) | (2u << 30); // global_addr[56:32] | type=2
  i32x8 g1;
  g1[0] = (int)(2u << 16);                               // data_size=2 -> 4 bytes
  g1[1] = (int)((remaining_elems & 0xFFFFu) << 16);      // tensor_dim0[15:0]
  g1[2] = (int)((remaining_elems >> 16) | (1u << 16));   // tensor_dim0[31:16] | tensor_dim1=1
  g1[3] = (int)((tile_elems & 0xFFFFu) << 16);           // tile_dim0
  g1[4] = 1;                                             // tile_dim1=1, tile_dim2=0
  g1[5] = (int)remaining_elems;                          // tensor_dim0_stride (unused, 1 row)
  g1[6] = 0;
  g1[7] = 0;
  i32x4 gz4 = {0, 0, 0, 0};
  i32x8 gz8 = {0, 0, 0, 0, 0, 0, 0, 0};
  __builtin_amdgcn_tensor_load_to_lds(g0, g1, gz4, gz4, gz8, 0);
}

// ---------------------------------------------------------------------------
// Elementwise helpers (float4-vectorized, memory-bound)
// ---------------------------------------------------------------------------
__global__ void k_zero4(float4* __restrict__ p, long long n4) {
  long long i = (long long)blockIdx.x * blockDim.x + threadIdx.x;
  if (i < n4) p[i] = make_float4(0.f, 0.f, 0.f, 0.f);
}

__global__ void k_deg(const int* __restrict__ col, float* __restrict__ deg, int E) {
  int e = blockIdx.x * blockDim.x + threadIdx.x;
  if (e < E) atomicAdd(&deg[col[e]], 1.0f);
}

__global__ void k_dinv(float* __restrict__ deg, int N) {
  int i = blockIdx.x * blockDim.x + threadIdx.x;
  if (i < N) {
    float d = deg[i];
    deg[i] = (d > 0.f) ? rsqrtf(d) : 0.f;   // in place: deg -> deg^-1/2
  }
}

__global__ void k_norm(const int* __restrict__ row, const int* __restrict__ col,
                       const float* __restrict__ dinv, float* __restrict__ nrm, int E) {
  int e = blockIdx.x * blockDim.x + threadIdx.x;
  if (e < E) nrm[e] = dinv[row[e]] * dinv[col[e]];
}

__global__ void k_init4(const float4* __restrict__ user, const float4* __restrict__ item,
                        float4* __restrict__ emb, float4* __restrict__ acc,
                        long long nu4, long long n4) {
  long long i = (long long)blockIdx.x * blockDim.x + threadIdx.x;
  if (i < n4) {
    float4 v = (i < nu4) ? user[i] : item[i - nu4];
    emb[i] = v;
    acc[i] = v;
  }
}

__global__ void k_add4(float4* __restrict__ acc, const float4* __restrict__ x, long long n4) {
  long long i = (long long)blockIdx.x * blockDim.x + threadIdx.x;
  if (i < n4) {
    float4 a = acc[i], b = x[i];
    acc[i] = make_float4(a.x + b.x, a.y + b.y, a.z + b.z, a.w + b.w);
  }
}

__global__ void k_scale4(float4* __restrict__ p, long long n4, float s) {
  long long i = (long long)blockIdx.x * blockDim.x + threadIdx.x;
  if (i < n4) {
    float4 a = p[i];
    p[i] = make_float4(a.x * s, a.y * s, a.z * s, a.w * s);
  }
}

// ---------------------------------------------------------------------------
// Scatter layer: nxt[col] += norm * cur[row]
// Each block owns LGC_CHUNK edges. Wave 0 DMAs row/col/norm into LDS via TDM,
// waits TENSORcnt==0, then all 8 waves process: 16 lanes per edge, float4 each
// (16 lanes x 4 floats = one 64-wide embedding row), 4 f32 global atomics.
//
// The compiler never sees a store to the shared arrays (the TDM engine writes
// them), so a runtime-unprovable dummy store keeps the loads from folding to
// undef while the arrays stay in addrspace(3) -> true ds_load_b32 with shared
// address register + immediate offsets.
// ---------------------------------------------------------------------------
__global__ void k_scatter(const int* __restrict__ rows, const int* __restrict__ cols,
                          const float* __restrict__ nrm, const float* __restrict__ cur,
                          float* __restrict__ nxt, int E) {
  __shared__ int   sh_row[LGC_CHUNK];
  __shared__ int   sh_col[LGC_CHUNK];
  __shared__ float sh_w[LGC_CHUNK];

  const int chunk0 = blockIdx.x * LGC_CHUNK;
  int count = E - chunk0;
  if (count <= 0) return;                       // uniform per block
  if (count > LGC_CHUNK) count = LGC_CHUNK;

  // Never true for our 1-D launches, but the compiler cannot prove it:
  // establishes the shared arrays as "may be stored to".
  if (blockDim.y == 7777u) {
    const int t = threadIdx.x & (LGC_CHUNK - 1);
    sh_row[t] = E;
    sh_col[t] = E;
    sh_w[t]   = (float)E;
  }

  if (threadIdx.x < 32) {                       // wave 0 only (uniform branch)
    const u32 rem = (u32)(E - chunk0);
    tdm_load_1d((u32)(unsigned long long)(const void*)sh_row, rows + chunk0, rem, LGC_CHUNK);
    tdm_load_1d((u32)(unsigned long long)(const void*)sh_col, cols + chunk0, rem, LGC_CHUNK);
    tdm_load_1d((u32)(unsigned long long)(const void*)sh_w,   nrm  + chunk0, rem, LGC_CHUNK);
    __builtin_amdgcn_s_wait_tensorcnt(0);
  }
  __syncthreads();

  const int lane16 = threadIdx.x & 15;          // which float4 of the 64-wide row
  const int slot   = threadIdx.x >> 4;          // 0..15: edge slot within block

  for (int e = slot; e < count; e += 16) {
    const int   r = sh_row[e];
    const int   c = sh_col[e];
    const float w = sh_w[e];

    // Prefetch next iteration's (random) source row while this one is in flight.
    const int en = e + 16;
    if (en < count) {
      __builtin_prefetch(cur + (size_t)sh_row[en] * 64 + lane16 * 4, 0, 1);
    }

    const float4 v = *(const float4*)(cur + (size_t)r * 64 + lane16 * 4);
    float* dst = nxt + (size_t)c * 64 + lane16 * 4;
    atomicAdd(dst + 0, v.x * w);
    atomicAdd(dst + 1, v.y * w);
    atomicAdd(dst + 2, v.z * w);
    atomicAdd(dst + 3, v.w * w);
  }
}

// ---------------------------------------------------------------------------
// Host-side orchestration
// ---------------------------------------------------------------------------
extern "C" void kernel_launch(void* const* d_in, const int* in_sizes, int n_in,
                              void* d_out, int out_size, void* d_ws, size_t ws_size,
                              hipStream_t stream) {
  const int*   edges = (const int*)d_in[0];
  const float* user  = (const float*)d_in[1];
  const float* item  = (const float*)d_in[2];

  const int E  = in_sizes[0] / 2;        // 4,000,000
  const int NU = in_sizes[1] / 64;       // 100,000
  const int NI = in_sizes[2] / 64;       // 50,000
  const int N  = NU + NI;                // 150,000

  const int* rows = edges;
  const int* cols = edges + E;

  float* acc = (float*)d_out;            // running layer sum lives in d_out

  // Workspace layout (floats): deg/dinv [N] | norm [E] | embA [N*64] | embB [N*64]
  float* deg  = (float*)d_ws;
  float* nrm  = deg + N;
  float* embA = nrm + E;
  float* embB = embA + (size_t)N * 64;

  const long long n4  = (long long)N * 64 / 4;   // emb table in float4s
  const long long nu4 = (long long)NU * 64 / 4;

  const int B = 256;
  const int gN    = (N + B - 1) / B;
  const int gE    = (E + B - 1) / B;
  const int gN4   = (int)((n4 + B - 1) / B);
  const int gScat = (E + LGC_CHUNK - 1) / LGC_CHUNK;

  // degree -> dinv -> per-edge norm (amortized over all layers)
  k_zero4<<<(int)(((long long)N / 4 + B - 1) / B), B, 0, stream>>>((float4*)deg, N / 4);
  k_deg<<<gE, B, 0, stream>>>(cols, deg, E);
  k_dinv<<<gN, B, 0, stream>>>(deg, N);
  k_norm<<<gE, B, 0, stream>>>(rows, cols, deg, nrm, E);

  // emb = concat(user, item); acc = emb
  k_init4<<<gN4, B, 0, stream>>>((const float4*)user, (const float4*)item,
                                 (float4*)embA, (float4*)acc, nu4, n4);

  float* cur = embA;
  float* nxt = embB;
  for (int layer = 0; layer < 4; ++layer) {
    k_zero4<<<gN4, B, 0, stream>>>((float4*)nxt, n4);
    k_scatter<<<gScat, SCATTER_BLOCK, 0, stream>>>(rows, cols, nrm, cur, nxt, E);
    k_add4<<<gN4, B, 0, stream>>>((float4*)acc, (const float4*)nxt, n4);
    float* t = cur; cur = nxt; nxt = t;
  }

  // acc / (L+1)^2 = acc / 25
  k_scale4<<<gN4, B, 0, stream>>>((float4*)acc, n4, 1.0f / 25.0f);
}